// GP_43336220017040
// MI455X (gfx1250) — compile-verified
//
#include <hip/hip_runtime.h>
#include <math.h>

typedef float v2f __attribute__((ext_vector_type(2)));
typedef float v8f __attribute__((ext_vector_type(8)));

#define N_TRAIN 4096
#define N_TEST  2048
#define DIMS    16
#define LDK     4096
#define LDW     2112   // 2048 test cols + 1 y col, padded to a multiple of 64
#define NB      64     // Cholesky block size

// ---------- WMMA helpers (V_WMMA_F32_16X16X4_F32, wave32) ----------

__device__ __forceinline__ v8f wmma4(v2f a, v2f b, v8f c) {
    // 8 args: (neg_a, A, neg_b, B, c_mod, C, reuse_a, reuse_b); neg must be 0 for f32
    return __builtin_amdgcn_wmma_f32_16x16x4_f32(false, a, false, b, (short)0, c, false, false);
}

// A fragment (16 rows x 4 k) from row-major matrix; also serves as B fragment
// when B = M^T for row-major M (gram-style). Contiguous float2 per lane.
__device__ __forceinline__ v2f ld_frag(const float* __restrict__ p, int row0, int ld, int k0) {
    int lane = threadIdx.x & 31;
    const float* q = p + (size_t)(row0 + (lane & 15)) * ld + k0 + ((lane >> 4) << 1);
    return *(const v2f*)q;
}

// B fragment (4 k x 16 cols) from a genuinely row-major K x N matrix (strided loads).
__device__ __forceinline__ v2f ld_fragB(const float* __restrict__ p, int k0, int ld, int col0) {
    int lane = threadIdx.x & 31;
    int c = col0 + (lane & 15);
    int k = k0 + ((lane >> 4) << 1);
    v2f b;
    b.x = p[(size_t)k * ld + c];
    b.y = p[(size_t)(k + 1) * ld + c];
    return b;
}

// C -= acc  (16x16 tile, f32 C/D layout)
__device__ __forceinline__ void store_sub(float* __restrict__ C, int ld, int r0, int c0, v8f acc) {
    int lane = threadIdx.x & 31;
    int col  = c0 + (lane & 15);
    int rb   = r0 + ((lane >> 4) << 3);
#pragma unroll
    for (int r = 0; r < 8; ++r) {
        size_t off = (size_t)(rb + r) * ld + col;
        C[off] = C[off] - acc[r];
    }
}

// ---------- 1. scale inputs, squared norms ----------

__global__ void k_prep(const float* __restrict__ tr, const float* __restrict__ te,
                       const float* __restrict__ ll2,
                       float* __restrict__ xs, float* __restrict__ ts,
                       float* __restrict__ sqx, float* __restrict__ sqt) {
    int i = blockIdx.x * blockDim.x + threadIdx.x;
    float sc[DIMS];
#pragma unroll
    for (int d = 0; d < DIMS; ++d) sc[d] = rsqrtf(2.0f * expf(ll2[d]));
    if (i < N_TRAIN) {
        float acc = 0.f;
#pragma unroll
        for (int d = 0; d < DIMS; ++d) {
            float v = tr[i * DIMS + d] * sc[d];
            xs[i * DIMS + d] = v;
            acc += v * v;
        }
        sqx[i] = acc;
    }
    if (i < N_TEST) {
        float acc = 0.f;
#pragma unroll
        for (int d = 0; d < DIMS; ++d) {
            float v = te[i * DIMS + d] * sc[d];
            ts[i * DIMS + d] = v;
            acc += v * v;
        }
        sqt[i] = acc;
    }
}

// ---------- 2. Knn lower triangle: sf2*exp(-max(d2,0)) + sn2*I ----------

__global__ void k_knn(const float* __restrict__ xs, const float* __restrict__ sqx,
                      const float* __restrict__ lsf2, const float* __restrict__ lsn2,
                      float* __restrict__ K) {
    int wave = threadIdx.x >> 5;
    int lane = threadIdx.x & 31;
    int rt = blockIdx.y * 8 + wave;       // 16-row tile index
    int ct = blockIdx.x;                  // 16-col tile index
    if (ct > rt) return;                  // wave-uniform: EXEC stays all-ones for WMMA
    int r0 = rt * 16, c0 = ct * 16;
    v8f acc = {};
#pragma unroll
    for (int k = 0; k < DIMS; k += 4) {
        v2f a = ld_frag(xs, r0, DIMS, k);
        v2f b = ld_frag(xs, c0, DIMS, k);   // B = xs^T
        acc = wmma4(a, b, acc);
    }
    float sf2 = expf(lsf2[0]);
    float sn2 = expf(lsn2[0]);
    int col = c0 + (lane & 15);
    int rb  = r0 + ((lane >> 4) << 3);
    float sqc = sqx[col];
#pragma unroll
    for (int r = 0; r < 8; ++r) {
        int row = rb + r;
        float d2 = fmaxf(sqx[row] + sqc - 2.0f * acc[r], 0.0f);
        float v = sf2 * expf(-d2);
        if (row == col) v += sn2;
        K[(size_t)row * LDK + col] = v;
    }
}

// ---------- 3. Kstar -> W[:, 0:2048] ----------

__global__ void k_kstar(const float* __restrict__ xs, const float* __restrict__ ts,
                        const float* __restrict__ sqx, const float* __restrict__ sqt,
                        const float* __restrict__ lsf2, float* __restrict__ W) {
    int wave = threadIdx.x >> 5;
    int lane = threadIdx.x & 31;
    int r0 = (blockIdx.y * 8 + wave) * 16;
    int c0 = blockIdx.x * 16;
    v8f acc = {};
#pragma unroll
    for (int k = 0; k < DIMS; k += 4) {
        v2f a = ld_frag(xs, r0, DIMS, k);
        v2f b = ld_frag(ts, c0, DIMS, k);   // B = ts^T
        acc = wmma4(a, b, acc);
    }
    float sf2 = expf(lsf2[0]);
    int col = c0 + (lane & 15);
    int rb  = r0 + ((lane >> 4) << 3);
    float sqc = sqt[col];
#pragma unroll
    for (int r = 0; r < 8; ++r) {
        int row = rb + r;
        float d2 = fmaxf(sqx[row] + sqc - 2.0f * acc[r], 0.0f);
        W[(size_t)row * LDW + col] = sf2 * expf(-d2);
    }
}

// ---------- 4. W[:, 2048] = y ; W[:, 2049:] = 0 ----------

__global__ void k_wfill(const float* __restrict__ y, float* __restrict__ W) {
    int i = blockIdx.x * blockDim.x + threadIdx.x;
    if (i < N_TRAIN) {
        W[(size_t)i * LDW + N_TEST] = y[i];
#pragma unroll
        for (int j = N_TEST + 1; j < LDW; ++j) W[(size_t)i * LDW + j] = 0.0f;
    }
}

// ---------- 5a. Cholesky of 64x64 diagonal block (one workgroup) ----------

__global__ void k_potrf(float* __restrict__ K, int kb) {
    __shared__ float L[NB * (NB + 1)];
    int t0 = kb * NB;
    for (int idx = threadIdx.x; idx < NB * NB; idx += blockDim.x) {
        int i = idx >> 6, c = idx & 63;
        L[i * (NB + 1) + c] = K[(size_t)(t0 + i) * LDK + t0 + c];
    }
    __syncthreads();
    for (int j = 0; j < NB; ++j) {
        if (threadIdx.x == 0) L[j * (NB + 1) + j] = sqrtf(L[j * (NB + 1) + j]);
        __syncthreads();
        float dinv = 1.0f / L[j * (NB + 1) + j];
        for (int i = j + 1 + (int)threadIdx.x; i < NB; i += blockDim.x)
            L[i * (NB + 1) + j] *= dinv;
        __syncthreads();
        for (int idx = threadIdx.x; idx < NB * NB; idx += blockDim.x) {
            int i = idx >> 6, c = idx & 63;
            if (c > j && c <= i)
                L[i * (NB + 1) + c] -= L[i * (NB + 1) + j] * L[c * (NB + 1) + j];
        }
        __syncthreads();
    }
    for (int idx = threadIdx.x; idx < NB * NB; idx += blockDim.x) {
        int i = idx >> 6, c = idx & 63;
        if (c <= i) K[(size_t)(t0 + i) * LDK + t0 + c] = L[i * (NB + 1) + c];
    }
}

// ---------- 5b. Panel TRSM: rows below diag, solve X * Lkk^T = A in-place ----------

__global__ void k_trsm_panel(float* __restrict__ K, int kb) {
    __shared__ float L[NB * (NB + 1)];
    int t0 = kb * NB;
    for (int idx = threadIdx.x; idx < NB * NB; idx += blockDim.x) {
        int i = idx >> 6, c = idx & 63;
        L[i * (NB + 1) + c] = K[(size_t)(t0 + i) * LDK + t0 + c];
    }
    __syncthreads();
    int r = t0 + NB + blockIdx.x * blockDim.x + threadIdx.x;
    if (r >= N_TRAIN) return;
    float* __restrict__ row = K + (size_t)r * LDK + t0;
    for (int c = 0; c < NB; ++c) {
        float v = row[c];
        for (int m = 0; m < c; ++m) v -= row[m] * L[c * (NB + 1) + m];
        row[c] = v / L[c * (NB + 1) + c];
    }
}

// ---------- 5c. SYRK trailing update: C -= P P^T (WMMA, 32x32 per wave) ----------

__global__ void k_syrk(float* __restrict__ K, int kb) {
    int t0 = (kb + 1) * NB;
    int wave = threadIdx.x >> 5;
    int r0 = t0 + blockIdx.y * 128 + (wave >> 1) * 32;
    int c0 = t0 + blockIdx.x * 64 + (wave & 1) * 32;
    if (r0 >= N_TRAIN || c0 >= N_TRAIN) return;  // wave-uniform
    if (c0 > r0 + 31) return;                    // strictly-upper tile: skip
    int pc = kb * NB;
    v8f c00 = {}, c01 = {}, c10 = {}, c11 = {};
#pragma unroll 4
    for (int k = 0; k < NB; k += 4) {
        v2f a0 = ld_frag(K, r0,      LDK, pc + k);
        v2f a1 = ld_frag(K, r0 + 16, LDK, pc + k);
        v2f b0 = ld_frag(K, c0,      LDK, pc + k);  // B = P^T
        v2f b1 = ld_frag(K, c0 + 16, LDK, pc + k);
        c00 = wmma4(a0, b0, c00);
        c01 = wmma4(a0, b1, c01);
        c10 = wmma4(a1, b0, c10);
        c11 = wmma4(a1, b1, c11);
    }
    store_sub(K, LDK, r0,      c0,      c00);
    store_sub(K, LDK, r0,      c0 + 16, c01);
    store_sub(K, LDK, r0 + 16, c0,      c10);
    store_sub(K, LDK, r0 + 16, c0 + 16, c11);
}

// ---------- 6a. Diag TRSM on W: solve Lkk * X = B, one thread per RHS column ----------

__global__ void k_trsmW(const float* __restrict__ K, float* __restrict__ W, int kb) {
    __shared__ float L[NB * (NB + 1)];
    int t0 = kb * NB;
    for (int idx = threadIdx.x; idx < NB * NB; idx += blockDim.x) {
        int i = idx >> 6, c = idx & 63;
        L[i * (NB + 1) + c] = K[(size_t)(t0 + i) * LDK + t0 + c];
    }
    __syncthreads();
    int j = blockIdx.x * blockDim.x + threadIdx.x;
    if (j >= LDW) return;
    float* __restrict__ col = W + (size_t)t0 * LDW + j;
    for (int i = 0; i < NB; ++i) {
        float v = col[(size_t)i * LDW];
        for (int m = 0; m < i; ++m) v -= L[i * (NB + 1) + m] * col[(size_t)m * LDW];
        col[(size_t)i * LDW] = v / L[i * (NB + 1) + i];
    }
}

// ---------- 6b. GEMM update: W[rem,:] -= Lpanel @ X  (WMMA, 32x32 per wave) ----------

__global__ void k_gemmW(const float* __restrict__ K, float* __restrict__ W, int kb) {
    int t0 = (kb + 1) * NB;
    int wave = threadIdx.x >> 5;
    int r0 = t0 + blockIdx.y * 128 + (wave >> 1) * 32;
    int c0 = blockIdx.x * 64 + (wave & 1) * 32;
    if (r0 >= N_TRAIN) return;  // wave-uniform
    int pc = kb * NB;           // L panel columns
    int xr = kb * NB;           // solved X rows in W
    v8f c00 = {}, c01 = {}, c10 = {}, c11 = {};
#pragma unroll 4
    for (int k = 0; k < NB; k += 4) {
        v2f a0 = ld_frag(K, r0,      LDK, pc + k);
        v2f a1 = ld_frag(K, r0 + 16, LDK, pc + k);
        v2f b0 = ld_fragB(W, xr + k, LDW, c0);
        v2f b1 = ld_fragB(W, xr + k, LDW, c0 + 16);
        c00 = wmma4(a0, b0, c00);
        c01 = wmma4(a0, b1, c01);
        c10 = wmma4(a1, b0, c10);
        c11 = wmma4(a1, b1, c11);
    }
    store_sub(W, LDW, r0,      c0,      c00);
    store_sub(W, LDW, r0,      c0 + 16, c01);
    store_sub(W, LDW, r0 + 16, c0,      c10);
    store_sub(W, LDW, r0 + 16, c0 + 16, c11);
}

// ---------- 7. finalize: mean = V^T Lslashy ; var = sf2 + sn2 - colsum(V*V) ----------

__global__ void k_finalize(const float* __restrict__ W,
                           const float* __restrict__ lsf2, const float* __restrict__ lsn2,
                           float* __restrict__ out) {
    int j = blockIdx.x * blockDim.x + threadIdx.x;
    if (j >= N_TEST) return;
    float mean = 0.f, ss = 0.f;
    for (int i = 0; i < N_TRAIN; ++i) {
        float v  = W[(size_t)i * LDW + j];
        float ls = W[(size_t)i * LDW + N_TEST];
        mean += v * ls;
        ss   += v * v;
    }
    float sf2 = expf(lsf2[0]);
    float sn2 = expf(lsn2[0]);
    out[j] = mean;
    out[N_TEST + j] = sf2 + sn2 - ss;
}

// ---------- host ----------

extern "C" void kernel_launch(void* const* d_in, const int* in_sizes, int n_in,
                              void* d_out, int out_size, void* d_ws, size_t ws_size,
                              hipStream_t stream) {
    const float* train = (const float*)d_in[0];
    const float* y     = (const float*)d_in[1];
    const float* test  = (const float*)d_in[2];
    const float* lsf2  = (const float*)d_in[3];
    const float* ll2   = (const float*)d_in[4];
    const float* lsn2  = (const float*)d_in[5];

    char* ws = (char*)d_ws;
    float* xs  = (float*)(ws);                         // 4096*16*4   = 256 KiB
    float* ts  = (float*)(ws + (256u << 10));          // 2048*16*4   = 128 KiB
    float* sqx = (float*)(ws + (384u << 10));          // 16 KiB
    float* sqt = (float*)(ws + (400u << 10));          // 8 KiB
    float* K   = (float*)(ws + (512u << 10));          // 4096*4096*4 = 64 MiB
    float* W   = (float*)(ws + (512u << 10) + (size_t)LDK * N_TRAIN * 4);  // 4096*2112*4

    float* out = (float*)d_out;

    k_prep <<<16, 256, 0, stream>>>(train, test, ll2, xs, ts, sqx, sqt);
    k_knn  <<<dim3(N_TRAIN / 16, N_TRAIN / 128), 256, 0, stream>>>(xs, sqx, lsf2, lsn2, K);
    k_kstar<<<dim3(N_TEST / 16, N_TRAIN / 128), 256, 0, stream>>>(xs, ts, sqx, sqt, lsf2, W);
    k_wfill<<<16, 256, 0, stream>>>(y, W);

    // Blocked right-looking Cholesky of K (lower)
    for (int kb = 0; kb < N_TRAIN / NB; ++kb) {
        k_potrf<<<1, 256, 0, stream>>>(K, kb);
        int rem = N_TRAIN - (kb + 1) * NB;
        if (rem > 0) {
            k_trsm_panel<<<(rem + 255) / 256, 256, 0, stream>>>(K, kb);
            k_syrk<<<dim3((rem + 63) / 64, (rem + 127) / 128), 256, 0, stream>>>(K, kb);
        }
    }

    // Blocked forward solve: W = L^{-1} [Kstar | y | 0...]
    for (int kb = 0; kb < N_TRAIN / NB; ++kb) {
        k_trsmW<<<(LDW + 255) / 256, 256, 0, stream>>>(K, W, kb);
        int rem = N_TRAIN - (kb + 1) * NB;
        if (rem > 0)
            k_gemmW<<<dim3(LDW / 64, (rem + 127) / 128), 256, 0, stream>>>(K, W, kb);
    }

    k_finalize<<<(N_TEST + 255) / 256, 256, 0, stream>>>(W, lsf2, lsn2, out);
}